// GNNChildEncoder_16681652978505
// MI455X (gfx1250) — compile-verified
//
#include <hip/hip_runtime.h>

typedef float v2f __attribute__((ext_vector_type(2)));
typedef float v4f __attribute__((ext_vector_type(4)));
typedef float v8f __attribute__((ext_vector_type(8)));

#define HID   128
#define MAXC  65536
#define NEDGE 1048576

// C[M x 128] = op(A[M x K] @ B[K x 128]); one block = 64 rows x 128 cols,
// 8 waves, wave w owns cols [16w,16w+16), 4 stacked 16x16 WMMA tiles along M.
// WMMA f32 16x16x4 fragment mapping (ISA 7.12.2):
//   A: lane&15 = M, (lane>>4)*2+v = K       (v2f per lane)
//   B: lane&15 = N, (lane>>4)*2+v = K       (mirrored)
//   C/D: vgpr r = row r (lanes 0-15) / row r+8 (lanes 16-31), lane&15 = N
__global__ __launch_bounds__(256)
void gemm_wmma_kernel(const float* __restrict__ A, int lda,
                      const float* __restrict__ B,        // K x 128 row-major
                      float* __restrict__ C,
                      const float* __restrict__ bias,     // may be null
                      const float* __restrict__ rowscale, // may be null (per-row)
                      int K, int do_relu)
{
    const int wave = threadIdx.x >> 5;
    const int lane = threadIdx.x & 31;
    const int col  = (wave << 4) | (lane & 15);
    const int koff = (lane >> 4) << 1;              // 0 or 2
    const int m0   = blockIdx.x * 64 + (lane & 15);

    const float* Arow[4];
#pragma unroll
    for (int t = 0; t < 4; ++t)
        Arow[t] = A + (size_t)(m0 + t * 16) * lda;

    v8f acc[4];
#pragma unroll
    for (int t = 0; t < 4; ++t)
        acc[t] = (v8f){0.f, 0.f, 0.f, 0.f, 0.f, 0.f, 0.f, 0.f};

    for (int k = 0; k < K; k += 4) {
        v2f b;
        b.x = B[(k + koff) * HID + col];
        b.y = B[(k + koff + 1) * HID + col];
#pragma unroll
        for (int t = 0; t < 4; ++t) {
            v2f a = *(const v2f*)(Arow[t] + k + koff);
            acc[t] = __builtin_amdgcn_wmma_f32_16x16x4_f32(
                false, a, false, b, (short)0, acc[t], false, false);
        }
    }

    const int rowAdd = (lane >> 4) << 3;            // 0 or 8
#pragma unroll
    for (int t = 0; t < 4; ++t) {
#pragma unroll
        for (int r = 0; r < 8; ++r) {
            int row = blockIdx.x * 64 + t * 16 + rowAdd + r;
            float v = acc[t][r];
            if (bias)     v += bias[col];
            if (do_relu)  v = fmaxf(v, 0.f);
            if (rowscale) v *= rowscale[row];
            C[(size_t)row * HID + col] = v;
        }
    }
}

__global__ __launch_bounds__(256)
void zero_kernel(float* __restrict__ p, int n4)
{
    int i = blockIdx.x * blockDim.x + threadIdx.x;
    if (i < n4) ((v4f*)p)[i] = (v4f){0.f, 0.f, 0.f, 0.f};
}

// One wave per edge; lane handles 4 columns (32*4 = 128).
// out = relu(y1[src] + y2[dst] + ef@W3 + b); scatter-add into xnew[src].
__global__ __launch_bounds__(256)
void edge_kernel(const float* __restrict__ y1, const float* __restrict__ y2,
                 const float* __restrict__ ef, const int* __restrict__ eidx,
                 const float* __restrict__ W3, const float* __restrict__ bedge,
                 float* __restrict__ xnew)
{
    const int e    = blockIdx.x * 8 + (threadIdx.x >> 5);
    const int lane = threadIdx.x & 31;
    const int c    = lane << 2;

    const int s = eidx[2 * e];
    const int d = eidx[2 * e + 1];

    v4f acc = *(const v4f*)(bedge + c);
#pragma unroll
    for (int k = 0; k < 8; ++k) {
        float w  = ef[(size_t)e * 8 + k];
        v4f  w3 = *(const v4f*)(W3 + k * HID + c);
        acc += w * w3;
    }
    v4f a = *(const v4f*)(y1 + (size_t)s * HID + c);
    v4f b = *(const v4f*)(y2 + (size_t)d * HID + c);
    acc += a + b;

    float* dst = xnew + (size_t)s * HID + c;
    unsafeAtomicAdd(dst + 0, fmaxf(acc.x, 0.f));
    unsafeAtomicAdd(dst + 1, fmaxf(acc.y, 0.f));
    unsafeAtomicAdd(dst + 2, fmaxf(acc.z, 0.f));
    unsafeAtomicAdd(dst + 3, fmaxf(acc.w, 0.f));
}

__global__ __launch_bounds__(128)
void colmax_part_kernel(const float* __restrict__ x, float* __restrict__ partial,
                        int rowsPerBlock)
{
    int t = threadIdx.x;
    size_t r0 = (size_t)blockIdx.x * rowsPerBlock;
    float m = -__builtin_inff();
    for (int r = 0; r < rowsPerBlock; ++r)
        m = fmaxf(m, x[(r0 + r) * HID + t]);
    partial[blockIdx.x * HID + t] = m;
}

__global__ __launch_bounds__(128)
void colmax_final_kernel(const float* __restrict__ partial, int nParts,
                         float* __restrict__ outv)
{
    int t = threadIdx.x;
    float m = -__builtin_inff();
    for (int p = 0; p < nParts; ++p)
        m = fmaxf(m, partial[p * HID + t]);
    outv[t] = m;
}

__global__ __launch_bounds__(128)
void parent_kernel(const float* __restrict__ mvec, const float* __restrict__ Wp,
                   const float* __restrict__ bp, float* __restrict__ out)
{
    int n = threadIdx.x;
    float acc = bp[n];
    for (int k = 0; k < 384; ++k)
        acc = fmaf(mvec[k], Wp[k * HID + n], acc);
    out[n] = fmaxf(acc, 0.f);
}

extern "C" void kernel_launch(void* const* d_in, const int* in_sizes, int n_in,
                              void* d_out, int out_size, void* d_ws, size_t ws_size,
                              hipStream_t stream)
{
    const float* CF     = (const float*)d_in[0];  // (65536,192)
    const float* exists = (const float*)d_in[1];  // (65536,1)
    const float* ef     = (const float*)d_in[2];  // (1M,8)
    const int*   eidx   = (const int*)  d_in[3];  // (1M,2)
    const float* Wc     = (const float*)d_in[4];  // (192,128)
    const float* bc     = (const float*)d_in[5];  // (128,)
    const float* We     = (const float*)d_in[6];  // (2,264,128)
    const float* be     = (const float*)d_in[7];  // (2,128)
    const float* Wp     = (const float*)d_in[8];  // (384,128)
    const float* bp     = (const float*)d_in[9];  // (128,)

    char* ws = (char*)d_ws;
    const size_t bufBytes = (size_t)MAXC * HID * sizeof(float);
    float* x       = (float*)(ws);                  // node feats / scatter target
    float* y1      = (float*)(ws + bufBytes);       // x @ W1
    float* y2      = (float*)(ws + 2 * bufBytes);   // x @ W2
    float* partial = (float*)(ws + 3 * bufBytes);   // 256 x 128
    float* mvec    = partial + 256 * HID;           // 384 (concat of col-maxes)

    // x = relu(CF @ Wc + bc) * exists
    gemm_wmma_kernel<<<MAXC / 64, 256, 0, stream>>>(CF, 192, Wc, x, bc, exists, 192, 1);
    colmax_part_kernel<<<256, 128, 0, stream>>>(x, partial, MAXC / 256);
    colmax_final_kernel<<<1, 128, 0, stream>>>(partial, 256, mvec);

    for (int it = 0; it < 2; ++it) {
        const float* W1 = We + (size_t)it * 264 * HID;  // rows   0..127
        const float* W2 = W1 + 128 * HID;               // rows 128..255
        const float* W3 = W1 + 256 * HID;               // rows 256..263
        gemm_wmma_kernel<<<MAXC / 64, 256, 0, stream>>>(x, HID, W1, y1, nullptr, nullptr, HID, 0);
        gemm_wmma_kernel<<<MAXC / 64, 256, 0, stream>>>(x, HID, W2, y2, nullptr, nullptr, HID, 0);
        zero_kernel<<<(MAXC * HID / 4) / 256, 256, 0, stream>>>(x, MAXC * HID / 4);
        edge_kernel<<<NEDGE / 8, 256, 0, stream>>>(y1, y2, ef, eidx, W3, be + it * HID, x);
        colmax_part_kernel<<<256, 128, 0, stream>>>(x, partial, MAXC / 256);
        colmax_final_kernel<<<1, 128, 0, stream>>>(partial, 256, mvec + (it + 1) * HID);
    }

    parent_kernel<<<1, 128, 0, stream>>>(mvec, Wp, bp, (float*)d_out);
}